// SeqCondAttentionBestNow_13477607375780
// MI455X (gfx1250) — compile-verified
//
#include <hip/hip_runtime.h>

typedef __attribute__((ext_vector_type(2))) float v2f;
typedef __attribute__((ext_vector_type(8))) float v8f;
typedef __attribute__((ext_vector_type(4))) int   v4i;

#define BB   2
#define LL   2048
#define DD   1024
#define KH   32          // num heads
#define HH   64          // head dim
#define DI   2048        // d_inner
#define ZW   4128        // 2*DI + KH
#define EPSF 1e-4f
#define NCH  16          // scan chunks along L
#define TCH  128         // tokens per chunk (NCH*TCH == LL)

#if defined(__has_builtin)
# if __has_builtin(__builtin_amdgcn_global_load_async_to_lds_b128)
#  define HAVE_ASYNC 1
# endif
#endif
#ifndef HAVE_ASYNC
# define HAVE_ASYNC 0
#endif

__device__ __forceinline__ void async_copy16(const float* gsrc, float* ldst) {
#if HAVE_ASYNC
    __builtin_amdgcn_global_load_async_to_lds_b128(
        (__attribute__((address_space(1))) v4i*)gsrc,
        (__attribute__((address_space(3))) v4i*)ldst, 0, 0);
#else
    const float4 v = *(const float4*)gsrc;
    *(float4*)ldst = v;
#endif
}

__device__ __forceinline__ void async_wait0() {
#if HAVE_ASYNC
    asm volatile("s_wait_asynccnt 0x0" ::: "memory");
#endif
}

__device__ __forceinline__ float softplusf(float x) {
    return (x > 20.f) ? x : log1pf(expf(x));
}

// ---------------- depthwise causal conv (KS=4) ----------------
__global__ void conv_dw(const float* __restrict__ x, const float* __restrict__ w,
                        float* __restrict__ out) {
    int id = blockIdx.x * 256 + threadIdx.x;           // over B*L*D
    int d  = id & (DD - 1);
    int bl = id >> 10;                                  // b*L + l
    int l  = bl & (LL - 1);
    float acc = 0.f;
#pragma unroll
    for (int j = 0; j < 4; ++j) {
        int li = l - 3 + j;
        if (li >= 0) acc += x[(size_t)(bl - l + li) * DD + d] * w[j * DD + d];
    }
    out[id] = acc;
}

// ---------------- generic fp32 WMMA GEMM: C[M,N] = A[M,K] @ B[K,N] ----------------
// M%128==0, N%32==0, K%32==0. Block = 128 thr (4 waves); block tile 128x32;
// wave tile 32x32 as four 16x16 accumulators (A/B fragments each reused twice).
// Double-buffered LDS staging via GLOBAL_LOAD_ASYNC_TO_LDS_B128 when available.
__global__ __launch_bounds__(128) void gemm_wmma_f32(const float* __restrict__ A,
                                                     const float* __restrict__ Bm,
                                                     float* __restrict__ C,
                                                     int M, int N, int Kd) {
    // pad 36: 16B-aligned rows, and 36*m mod 64 distinct for m=0..15 (36=4*9, 9 odd)
    __shared__ float As[2][128][36];
    __shared__ float Bs[2][32][36];
    const int tid  = threadIdx.x;
    const int lane = tid & 31;
    const int w    = tid >> 5;           // 0..3
    const int m0   = blockIdx.y * 128;
    const int n0   = blockIdx.x * 32;
    const int mloc = lane & 15;
    const int ksel = (lane >> 4) << 1;   // 0 for lanes 0-15, 2 for 16-31
    const int r0   = w * 32;
    v8f c00 = {}, c01 = {}, c10 = {}, c11 = {};

    const int nk = Kd >> 5;              // K blocks of 32

    // stage block 0 into buffer 0
    {
        const int k0 = 0;
#pragma unroll
        for (int p = 0; p < 8; ++p) {    // A tile 128x32 = 1024 float4
            int idx = tid + p * 128;
            int row = idx >> 3;
            int c4  = (idx & 7) << 2;
            async_copy16(A + (size_t)(m0 + row) * Kd + k0 + c4, &As[0][row][c4]);
        }
#pragma unroll
        for (int p = 0; p < 2; ++p) {    // B tile 32x32 = 256 float4
            int idx = tid + p * 128;
            int kk  = idx >> 3;
            int j4  = (idx & 7) << 2;
            async_copy16(Bm + (size_t)(k0 + kk) * N + n0 + j4, &Bs[0][kk][j4]);
        }
    }
    async_wait0();
    __syncthreads();

    for (int i = 0; i < nk; ++i) {
        const int bsel = i & 1;
        if (i + 1 < nk) {                // prefetch next K block into other buffer
            const int k0 = (i + 1) << 5;
#pragma unroll
            for (int p = 0; p < 8; ++p) {
                int idx = tid + p * 128;
                int row = idx >> 3;
                int c4  = (idx & 7) << 2;
                async_copy16(A + (size_t)(m0 + row) * Kd + k0 + c4, &As[bsel ^ 1][row][c4]);
            }
#pragma unroll
            for (int p = 0; p < 2; ++p) {
                int idx = tid + p * 128;
                int kk  = idx >> 3;
                int j4  = (idx & 7) << 2;
                async_copy16(Bm + (size_t)(k0 + kk) * N + n0 + j4, &Bs[bsel ^ 1][kk][j4]);
            }
        }
        const float (*Asb)[36] = As[bsel];
        const float (*Bsb)[36] = Bs[bsel];
#pragma unroll
        for (int kk = 0; kk < 32; kk += 4) {
            v2f a0, a1, b0, b1;
            a0.x = Asb[r0 + mloc][kk + ksel];
            a0.y = Asb[r0 + mloc][kk + ksel + 1];
            a1.x = Asb[r0 + 16 + mloc][kk + ksel];
            a1.y = Asb[r0 + 16 + mloc][kk + ksel + 1];
            b0.x = Bsb[kk + ksel][mloc];
            b0.y = Bsb[kk + ksel + 1][mloc];
            b1.x = Bsb[kk + ksel][16 + mloc];
            b1.y = Bsb[kk + ksel + 1][16 + mloc];
            c00 = __builtin_amdgcn_wmma_f32_16x16x4_f32(false, a0, false, b0, (short)0, c00, false, false);
            c01 = __builtin_amdgcn_wmma_f32_16x16x4_f32(false, a0, false, b1, (short)0, c01, false, false);
            c10 = __builtin_amdgcn_wmma_f32_16x16x4_f32(false, a1, false, b0, (short)0, c10, false, false);
            c11 = __builtin_amdgcn_wmma_f32_16x16x4_f32(false, a1, false, b1, (short)0, c11, false, false);
        }
        async_wait0();
        __syncthreads();
    }

    // C 16x16: VGPR i -> row i (lanes 0-15) / i+8 (lanes 16-31), col = lane&15
    const int rbase = m0 + r0 + ((lane >> 4) << 3);
#pragma unroll
    for (int i = 0; i < 8; ++i) {
        C[(size_t)(rbase + i) * N + n0 + mloc]           = c00[i];
        C[(size_t)(rbase + i) * N + n0 + 16 + mloc]      = c01[i];
        C[(size_t)(rbase + 16 + i) * N + n0 + mloc]      = c10[i];
        C[(size_t)(rbase + 16 + i) * N + n0 + 16 + mloc] = c11[i];
    }
}

// ---------------- scan phase 1: per-chunk sums ----------------
__global__ void scan_partials(const float* __restrict__ z, const float* __restrict__ decay,
                              const float* __restrict__ sscale, const float* __restrict__ theta,
                              float* __restrict__ pre, float* __restrict__ pim,
                              float* __restrict__ pden) {
    const int h = threadIdx.x;
    const int ch = blockIdx.x, k = blockIdx.y, b = blockIdx.z;
    const float slope = softplusf(decay[k]);
    const float sc    = sscale[k];
    const float th    = theta[k * HH + h];
    float sre = 0.f, sim = 0.f, sden = 0.f;
    for (int j = 0; j < TCH; ++j) {
        int l = ch * TCH + j;
        const float* zr = z + (size_t)(b * LL + l) * ZW;
        float pw = (softplusf(zr[2 * DI + k] * sc) + EPSF) * expf(-slope * (float)(LL - 1 - l));
        float xv = zr[k * HH + h];
        float s, c; sincosf(xv * th, &s, &c);
        sre += pw * c; sim += pw * s; sden += pw;
    }
    size_t o = ((size_t)((b * KH + k) * NCH + ch)) * HH + h;
    pre[o] = sre; pim[o] = sim;
    if (h == 0) pden[(b * KH + k) * NCH + ch] = sden;
}

// ---------------- scan phase 2: exclusive scan of chunk totals ----------------
__global__ void scan_offsets(float* __restrict__ pre, float* __restrict__ pim,
                             float* __restrict__ pden) {
    const int h = threadIdx.x;
    const int k = blockIdx.x, b = blockIdx.y;
    size_t base = ((size_t)(b * KH + k)) * NCH * HH + h;
    float r = 0.f;
    for (int ch = 0; ch < NCH; ++ch) { size_t o = base + (size_t)ch * HH; float t = pre[o]; pre[o] = r; r += t; }
    r = 0.f;
    for (int ch = 0; ch < NCH; ++ch) { size_t o = base + (size_t)ch * HH; float t = pim[o]; pim[o] = r; r += t; }
    if (h == 0) {
        float rd = 0.f;
        for (int ch = 0; ch < NCH; ++ch) { int o = (b * KH + k) * NCH + ch; float t = pden[o]; pden[o] = rd; rd += t; }
    }
}

// ---------------- scan phase 3: emit re/im = cumsum/den ----------------
__global__ void scan_emit(const float* __restrict__ z, const float* __restrict__ decay,
                          const float* __restrict__ sscale, const float* __restrict__ theta,
                          const float* __restrict__ pre, const float* __restrict__ pim,
                          const float* __restrict__ pden, float* __restrict__ R) {
    const int h = threadIdx.x;
    const int ch = blockIdx.x, k = blockIdx.y, b = blockIdx.z;
    const float slope = softplusf(decay[k]);
    const float sc    = sscale[k];
    const float th    = theta[k * HH + h];
    size_t o = ((size_t)((b * KH + k) * NCH + ch)) * HH + h;
    float are = pre[o], aim = pim[o];
    float aden = pden[(b * KH + k) * NCH + ch];
    for (int j = 0; j < TCH; ++j) {
        int l = ch * TCH + j;
        const float* zr = z + (size_t)(b * LL + l) * ZW;
        float pw = (softplusf(zr[2 * DI + k] * sc) + EPSF) * expf(-slope * (float)(LL - 1 - l));
        float xv = zr[k * HH + h];
        float s, c; sincosf(xv * th, &s, &c);
        aden += pw; are += pw * c; aim += pw * s;
        float inv = 1.0f / fmaxf(aden, EPSF);
        size_t ro = ((size_t)((b * LL + l) * KH + k)) * 128;
        R[ro + h]      = are * inv;   // re
        R[ro + 64 + h] = aim * inv;   // im
    }
}

// ---------------- RMS + mixing GEMM + SiLU gate ----------------
// g[t, h'] = rs[t] * sum_d R[t,d] * Wc[d,h'] * silu(gate), Wc = [ms_re*W_re ; ms_im*W_im]
// Block: 256 thr, 32 tokens x 64 cols, K=128. rs computed while staging A.
__global__ __launch_bounds__(256) void mix_gemm(const float* __restrict__ R,
                                                const float* __restrict__ Wre,
                                                const float* __restrict__ Wim,
                                                const float* __restrict__ ms,
                                                const float* __restrict__ z,
                                                float* __restrict__ g) {
    __shared__ float As[32][129];
    __shared__ float Bs[128][64];
    __shared__ float sq[32][8];
    __shared__ float rsA[32];
    const int tid = threadIdx.x;
    const int t0  = blockIdx.x * 32;

#pragma unroll
    for (int p = 0; p < 32; ++p) {                  // Wc: 128x64 with mix_scale folded
        int idx = tid + p * 256;
        int d = idx >> 6, hc = idx & 63;
        float wv = (d < 64) ? Wre[d * 64 + hc] : Wim[(d - 64) * 64 + hc];
        Bs[d][hc] = wv * ms[d];
    }
    {                                               // A: 32x128, sumsq on the fly
        int r = tid >> 3, q = tid & 7;
        float ssq = 0.f;
#pragma unroll
        for (int e = 0; e < 4; ++e) {
            int c = q * 16 + e * 4;
            const float4 v = *(const float4*)(R + (size_t)(t0 + r) * 128 + c);
            As[r][c] = v.x; As[r][c + 1] = v.y; As[r][c + 2] = v.z; As[r][c + 3] = v.w;
            ssq += v.x * v.x + v.y * v.y + v.z * v.z + v.w * v.w;
        }
        sq[r][q] = ssq;
    }
    __syncthreads();
    if (tid < 32) {
        float s = 0.f;
#pragma unroll
        for (int q = 0; q < 8; ++q) s += sq[tid][q];
        rsA[tid] = rsqrtf(s * (1.f / 128.f) + 1e-5f);
    }
    __syncthreads();

    const int lane = tid & 31, w = tid >> 5;
    const int r0   = (w & 1) * 16;
    const int cb   = (w >> 1) * 16;
    const int mloc = lane & 15;
    const int ksel = (lane >> 4) << 1;
    v8f c0 = {};
#pragma unroll
    for (int kk = 0; kk < 128; kk += 4) {
        v2f a, bb;
        a.x  = As[r0 + mloc][kk + ksel];
        a.y  = As[r0 + mloc][kk + ksel + 1];
        bb.x = Bs[kk + ksel][cb + mloc];
        bb.y = Bs[kk + ksel + 1][cb + mloc];
        c0 = __builtin_amdgcn_wmma_f32_16x16x4_f32(false, a, false, bb, (short)0, c0, false, false);
    }
    const int rb  = r0 + ((lane >> 4) << 3);
    const int col = cb + mloc;
#pragma unroll
    for (int i = 0; i < 8; ++i) {
        int row = rb + i;
        int t   = t0 + row;
        int k   = t & (KH - 1);
        int bl  = t >> 5;
        float gt  = z[(size_t)bl * ZW + DI + k * HH + col];
        float sig = 1.f / (1.f + expf(-gt));
        g[(size_t)bl * DI + k * HH + col] = c0[i] * rsA[row] * gt * sig;
    }
}

extern "C" void kernel_launch(void* const* d_in, const int* in_sizes, int n_in,
                              void* d_out, int out_size, void* d_ws, size_t ws_size,
                              hipStream_t stream) {
    const float* x      = (const float*)d_in[0];
    const float* conv_w = (const float*)d_in[1];
    const float* z_w    = (const float*)d_in[2];
    const float* decay  = (const float*)d_in[3];
    const float* sscale = (const float*)d_in[4];
    const float* theta  = (const float*)d_in[5];
    const float* Wre    = (const float*)d_in[6];
    const float* Wim    = (const float*)d_in[7];
    const float* ms     = (const float*)d_in[8];
    const float* out_w  = (const float*)d_in[9];
    float* out = (float*)d_out;

    float* ws    = (float*)d_ws;
    float* xconv = ws;                                    // B*L*D
    float* z     = xconv + (size_t)BB * LL * DD;          // B*L*4128
    float* R     = z + (size_t)BB * LL * ZW;              // B*L*K*128 (re||im)
    float* g     = R + (size_t)BB * LL * KH * 128;        // B*L*DI
    float* pre   = g + (size_t)BB * LL * DI;
    float* pim   = pre + (size_t)BB * KH * NCH * HH;
    float* pden  = pim + (size_t)BB * KH * NCH * HH;

    conv_dw<<<dim3(BB * LL * DD / 256), 256, 0, stream>>>(x, conv_w, xconv);
    gemm_wmma_f32<<<dim3(ZW / 32, BB * LL / 128), 128, 0, stream>>>(xconv, z_w, z, BB * LL, ZW, DD);
    scan_partials<<<dim3(NCH, KH, BB), HH, 0, stream>>>(z, decay, sscale, theta, pre, pim, pden);
    scan_offsets<<<dim3(KH, BB), HH, 0, stream>>>(pre, pim, pden);
    scan_emit<<<dim3(NCH, KH, BB), HH, 0, stream>>>(z, decay, sscale, theta, pre, pim, pden, R);
    mix_gemm<<<dim3(BB * LL * KH / 32), 256, 0, stream>>>(R, Wre, Wim, ms, z, g);
    gemm_wmma_f32<<<dim3(DD / 32, BB * LL / 128), 128, 0, stream>>>(g, out_w, out, BB * LL, DD, DI);
}